// PipelineRNN_90864328114407
// MI455X (gfx1250) — compile-verified
//
#include <hip/hip_runtime.h>

// PipelineRNN for MI455X (gfx1250, wave32).
// Phase 1: x_proj = x @ W_ih^T + b_ih + b_hh  (HBM-bound, fp32 WMMA 16x16x4)
// Phase 2: sequential scan h_t = tanh(xp_t + h @ W_hh^T):
//   - W_hh B-fragments hoisted to registers (loop-invariant over 2048 steps)
//   - 4 independent WMMA accumulation chains (K split 4x32) -> short critical path
//   - double-buffered h in LDS -> single barrier per step
//   - xp stream prefetched 2 steps ahead, outputs written in place in d_out

typedef __attribute__((ext_vector_type(2))) float v2f;
typedef __attribute__((ext_vector_type(4))) float v4f;
typedef __attribute__((ext_vector_type(8))) float v8f;

#define BB   256
#define TT   2048
#define KK   128   // I == H == 128
#define KPAD 132   // padded LDS row stride (floats): bank-conflict-free + 16B aligned
#define HSZ  (16 * KPAD)  // one h buffer

__device__ __forceinline__ float fast_tanh(float x) {
#if __has_builtin(__builtin_amdgcn_tanhf)
  return __builtin_amdgcn_tanhf(x);
#else
  float xc = fminf(fmaxf(x, -9.0f), 9.0f);
  float e  = __builtin_amdgcn_exp2f(xc * 2.8853900817779268f);
  return (e - 1.0f) * __builtin_amdgcn_rcpf(e + 1.0f);
#endif
}

// -------------------------------------------------------------------------
// Kernel 1: x_proj into d_out.  Grid: (B*T)/128 blocks, 256 threads (8 waves).
// -------------------------------------------------------------------------
__global__ __launch_bounds__(256) void rnn_xproj(
    const float* __restrict__ x, const float* __restrict__ w_ih,
    const float* __restrict__ b_ih, const float* __restrict__ b_hh,
    float* __restrict__ out)
{
  extern __shared__ float smem[];
  float* xs = smem;               // 128 x KPAD
  float* ws = smem + 128 * KPAD;  // 128 x KPAD  (W_ih, [n][k] row-major)

  const int tid  = threadIdx.x;
  const int wave = tid >> 5;
  const int lane = tid & 31;
  const int half = lane >> 4;
  const int lm   = lane & 15;
  const long m0  = (long)blockIdx.x * 128;

  for (int it = 0; it < 16; ++it) {
    int chunk = it * 256 + tid;
    int r = chunk >> 5, c4 = chunk & 31;
    v4f v = *(const v4f*)(w_ih + r * KK + c4 * 4);
    *(v4f*)(ws + r * KPAD + c4 * 4) = v;
  }
  for (int it = 0; it < 16; ++it) {
    int chunk = it * 256 + tid;
    int r = chunk >> 5, c4 = chunk & 31;
    v4f v = *(const v4f*)(x + (m0 + r) * KK + c4 * 4);
    *(v4f*)(xs + r * KPAD + c4 * 4) = v;
  }
  __syncthreads();

  v8f   c[8] = {};
  float bias[8];
#pragma unroll
  for (int nt = 0; nt < 8; ++nt) {
    int n = nt * 16 + lm;
    bias[nt] = b_ih[n] + b_hh[n];
  }

  const float* arow = xs + (wave * 16 + lm) * KPAD + 2 * half;
#pragma unroll 4
  for (int k0 = 0; k0 < KK; k0 += 4) {
    v2f a; a.x = arow[k0]; a.y = arow[k0 + 1];
#pragma unroll
    for (int nt = 0; nt < 8; ++nt) {
      const float* brow = ws + (nt * 16 + lm) * KPAD + k0 + 2 * half;
      v2f b; b.x = brow[0]; b.y = brow[1];
      c[nt] = __builtin_amdgcn_wmma_f32_16x16x4_f32(
          false, a, false, b, (short)0, c[nt], false, false);
    }
  }

#pragma unroll
  for (int nt = 0; nt < 8; ++nt) {
#pragma unroll
    for (int v = 0; v < 8; ++v) {
      long m = m0 + wave * 16 + v + 8 * half;
      out[m * KK + nt * 16 + lm] = c[nt][v] + bias[nt];
    }
  }
}

// -------------------------------------------------------------------------
// Kernel 2: recurrence, in place in d_out.  Grid: B/16 = 16 blocks, 256 thr.
// -------------------------------------------------------------------------
__global__ __launch_bounds__(256, 1) void rnn_scan(
    const float* __restrict__ w_hh, float* __restrict__ io)
{
  extern __shared__ float smem[];
  float* ws = smem;               // 128 x KPAD  (W_hh, [n][k])
  float* hs = smem + 128 * KPAD;  // 2 buffers of 16 x KPAD (double-buffered h)

  const int tid  = threadIdx.x;
  const int wave = tid >> 5;   // column tile
  const int lane = tid & 31;
  const int half = lane >> 4;
  const int lm   = lane & 15;
  const int b0   = blockIdx.x * 16;

  for (int it = 0; it < 16; ++it) {
    int chunk = it * 256 + tid;
    int r = chunk >> 5, c4 = chunk & 31;
    v4f v = *(const v4f*)(w_hh + r * KK + c4 * 4);
    *(v4f*)(ws + r * KPAD + c4 * 4) = v;
  }
  for (int i = tid; i < 2 * HSZ; i += 256) hs[i] = 0.0f;  // h_0 = 0
  __syncthreads();

  // Hoist the wave's entire W_hh B-panel into registers (invariant over t).
  // B frag chunk kc: lane -> N = lm, K = 4*kc + 2*half + {0,1}; B[k][n] = W[n][k]
  const float* bbase = ws + (wave * 16 + lm) * KPAD + 2 * half;
  v2f bf[32];
#pragma unroll
  for (int kc = 0; kc < 32; ++kc) {
    bf[kc].x = bbase[kc * 4];
    bf[kc].y = bbase[kc * 4 + 1];
  }

  float* pbase = io + (long)(b0 + 8 * half) * TT * KK + wave * 16 + lm;

  float pa[8], pb[8];  // xp pipeline: pa = step t, pb = step t+1
#pragma unroll
  for (int v = 0; v < 8; ++v) pa[v] = pbase[(long)v * TT * KK];
#pragma unroll
  for (int v = 0; v < 8; ++v) pb[v] = pbase[(long)v * TT * KK + KK];

  for (int t = 0; t < TT; ++t) {
    // 4 independent accumulation chains over K = 4 x 32; c0 seeded with xp_t
    v8f c0, c1 = {}, c2 = {}, c3 = {};
#pragma unroll
    for (int v = 0; v < 8; ++v) c0[v] = pa[v];
#pragma unroll
    for (int v = 0; v < 8; ++v) pa[v] = pb[v];
    if (t + 2 < TT) {
      long off = (long)(t + 2) * KK;
#pragma unroll
      for (int v = 0; v < 8; ++v) pb[v] = pbase[(long)v * TT * KK + off];
    }

    const float* ar = hs + (t & 1) * HSZ + lm * KPAD + 2 * half;  // A: h rows
#pragma unroll
    for (int kk = 0; kk < 32; kk += 4) {
      v2f a0; a0.x = ar[kk];      a0.y = ar[kk + 1];
      v2f a1; a1.x = ar[32 + kk]; a1.y = ar[32 + kk + 1];
      v2f a2; a2.x = ar[64 + kk]; a2.y = ar[64 + kk + 1];
      v2f a3; a3.x = ar[96 + kk]; a3.y = ar[96 + kk + 1];
      int kc = kk >> 2;
      c0 = __builtin_amdgcn_wmma_f32_16x16x4_f32(false, a0, false, bf[kc],      (short)0, c0, false, false);
      c1 = __builtin_amdgcn_wmma_f32_16x16x4_f32(false, a1, false, bf[kc + 8],  (short)0, c1, false, false);
      c2 = __builtin_amdgcn_wmma_f32_16x16x4_f32(false, a2, false, bf[kc + 16], (short)0, c2, false, false);
      c3 = __builtin_amdgcn_wmma_f32_16x16x4_f32(false, a3, false, bf[kc + 24], (short)0, c3, false, false);
    }
    c0 += c1; c2 += c3; c0 += c2;
#pragma unroll
    for (int v = 0; v < 8; ++v) c0[v] = fast_tanh(c0[v]);

    // write h_t: next-step LDS buffer + global output (overwrites consumed xp_t)
    float* hw = hs + ((t + 1) & 1) * HSZ;
    float* pt = pbase + (long)t * KK;
#pragma unroll
    for (int v = 0; v < 8; ++v) {
      hw[(v + 8 * half) * KPAD + wave * 16 + lm] = c0[v];
      pt[(long)v * TT * KK] = c0[v];
    }
    __syncthreads();  // step t writes visible; also fences step t's buf reads
                      // from step t+1's writes (different buffers each step)
  }
}

extern "C" void kernel_launch(void* const* d_in, const int* in_sizes, int n_in,
                              void* d_out, int out_size, void* d_ws, size_t ws_size,
                              hipStream_t stream) {
  (void)in_sizes; (void)n_in; (void)d_ws; (void)ws_size; (void)out_size;
  const float* x    = (const float*)d_in[0];
  const float* w_ih = (const float*)d_in[1];
  const float* w_hh = (const float*)d_in[2];
  const float* b_ih = (const float*)d_in[3];
  const float* b_hh = (const float*)d_in[4];
  float* out = (float*)d_out;

  size_t lds1 = (size_t)(128 * KPAD + 128 * KPAD) * sizeof(float);  // ~135 KB
  rnn_xproj<<<(BB * TT) / 128, 256, lds1, stream>>>(x, w_ih, b_ih, b_hh, out);

  size_t lds2 = (size_t)(128 * KPAD + 2 * HSZ) * sizeof(float);     // ~84 KB
  rnn_scan<<<BB / 16, 256, lds2, stream>>>(w_hh, out);
}